// SubspaceMethod_45930380263618
// MI455X (gfx1250) — compile-verified
//
#include <hip/hip_runtime.h>
#include <hip/hip_bf16.h>

typedef float v2f __attribute__((ext_vector_type(2)));
typedef float v8f __attribute__((ext_vector_type(8)));

#define B_TOT   2048
#define M_DIM   64
#define T_DIM   256
#define K_SRC   8
#define THRESH  0.3f
#define XS      68          // padded LDS row stride for X slices (conflict-free)
#define NSWEEP  6

// output layout (floats): signal (B,64,8 cplx) | noise (B,64,56 cplx) | src_est (B) | l_eig (1)
static constexpr size_t SIG_OFF  = 0;
static constexpr size_t NOI_OFF  = (size_t)B_TOT * 64 * 8 * 2;
static constexpr size_t SRC_OFF  = NOI_OFF + (size_t)B_TOT * 64 * 56 * 2;
static constexpr size_t LEIG_OFF = SRC_OFF + (size_t)B_TOT;

__global__ void leig_init_kernel(float* __restrict__ out) {
  if (threadIdx.x == 0 && blockIdx.x == 0) out[LEIG_OFF] = 0.0f;
}

// async 16B global -> LDS copy (GV mode), tracked on ASYNCcnt (no VGPR round-trip)
__device__ __forceinline__ void async_copy_b128(uint32_t lds_byte_addr,
                                                const void* gaddr) {
  asm volatile("global_load_async_to_lds_b128 %0, %1, off"
               :: "v"(lds_byte_addr), "v"((uint64_t)(uintptr_t)gaddr)
               : "memory");
}
__device__ __forceinline__ void wait_async0() {
  asm volatile("s_wait_asynccnt 0x0" ::: "memory");
}

__global__ __launch_bounds__(256) void subspace_kernel(
    const float* __restrict__ xre, const float* __restrict__ xim,
    float* __restrict__ out)
{
  extern __shared__ float smem[];
  // persistent partitions
  float* A_re = smem;            // 64*64
  float* A_im = smem + 4096;     // 64*64
  float* V_re = smem + 8192;     // 64*64
  float* V_im = smem + 12288;    // 64*64
  float* prm  = smem + 16384;    // c[32] zr[32] zi[32] ev[64] nrm[64] perm[64]
  float* cP  = prm;
  float* zrP = prm + 32;
  float* ziP = prm + 64;
  float* ev  = prm + 96;
  float* nrm = prm + 160;
  int*   perm = (int*)(prm + 224);
  // phase-1 staging buffers (overlap A/V region; dead before A is written)
  float* xr = smem;              // 64*XS = 4352
  float* xi = smem + 4352;       // 64*XS = 4352 (ends at 8704)

  const int tid  = threadIdx.x;
  const int b    = blockIdx.x;
  const int lane = tid & 31;
  const int wv   = tid >> 5;       // 0..7 (8 wave32)
  const int ln   = lane & 15;
  const int kh   = lane >> 4;      // K-half select for 16x16x4 fragments
  const int mi   = wv >> 1;        // M-tile (shared by both of this wave's tiles)
  const int ni0  = (2 * wv) & 3;   // N-tile of tile 0
  const int ni1  = ni0 + 1;        // N-tile of tile 1

  // ---------------- Phase 1: Rx = X X^H / T via WMMA ----------------
  v8f acc_re0 = {}, acc_im0 = {}, acc_n0 = {};
  v8f acc_re1 = {}, acc_im1 = {}, acc_n1 = {};

  const float* gre = xre + (size_t)b * M_DIM * T_DIM;
  const float* gim = xim + (size_t)b * M_DIM * T_DIM;

  const int lrow = tid >> 2;        // 0..63 : row loaded by this thread
  const int lq   = (tid & 3) * 16;  // 16-col quarter

  for (int s = 0; s < 4; ++s) {
    __syncthreads();                // previous slice fully consumed
#pragma unroll
    for (int j = 0; j < 4; ++j) {
      const int colo = lq + j * 4;
      async_copy_b128((uint32_t)(uintptr_t)(xr + lrow * XS + colo),
                      gre + lrow * T_DIM + s * 64 + colo);
      async_copy_b128((uint32_t)(uintptr_t)(xi + lrow * XS + colo),
                      gim + lrow * T_DIM + s * 64 + colo);
    }
    wait_async0();                  // drain this wave's async copies
    __syncthreads();                // make all waves' LDS writes visible
    for (int kk = 0; kk < 64; kk += 4) {
      const int co = kk + kh * 2;
      const v2f ar  = *(const v2f*)(xr + (mi  * 16 + ln) * XS + co);
      const v2f ai  = *(const v2f*)(xi + (mi  * 16 + ln) * XS + co);
      const v2f br0 = *(const v2f*)(xr + (ni0 * 16 + ln) * XS + co);
      const v2f bi0 = *(const v2f*)(xi + (ni0 * 16 + ln) * XS + co);
      const v2f br1 = *(const v2f*)(xr + (ni1 * 16 + ln) * XS + co);
      const v2f bi1 = *(const v2f*)(xi + (ni1 * 16 + ln) * XS + co);
      // Re += Xr Xr^T + Xi Xi^T ; Im(+) += Xi Xr^T ; Im(-) += Xr Xi^T
      acc_re0 = __builtin_amdgcn_wmma_f32_16x16x4_f32(false, ar, false, br0, (short)0, acc_re0, false, false);
      acc_re0 = __builtin_amdgcn_wmma_f32_16x16x4_f32(false, ai, false, bi0, (short)0, acc_re0, false, false);
      acc_im0 = __builtin_amdgcn_wmma_f32_16x16x4_f32(false, ai, false, br0, (short)0, acc_im0, false, false);
      acc_n0  = __builtin_amdgcn_wmma_f32_16x16x4_f32(false, ar, false, bi0, (short)0, acc_n0,  false, false);
      acc_re1 = __builtin_amdgcn_wmma_f32_16x16x4_f32(false, ar, false, br1, (short)0, acc_re1, false, false);
      acc_re1 = __builtin_amdgcn_wmma_f32_16x16x4_f32(false, ai, false, bi1, (short)0, acc_re1, false, false);
      acc_im1 = __builtin_amdgcn_wmma_f32_16x16x4_f32(false, ai, false, br1, (short)0, acc_im1, false, false);
      acc_n1  = __builtin_amdgcn_wmma_f32_16x16x4_f32(false, ar, false, bi1, (short)0, acc_n1,  false, false);
    }
  }
  __syncthreads();  // staging buffers dead; safe to overwrite with A

  const float scale = 1.0f / (float)T_DIM;
#pragma unroll
  for (int v = 0; v < 8; ++v) {
    const int row = mi * 16 + v + 8 * kh;   // C layout: lanes 16-31 hold M+8
    const int c0  = ni0 * 16 + ln;
    const int c1  = ni1 * 16 + ln;
    A_re[row * 64 + c0] = acc_re0[v] * scale;
    A_im[row * 64 + c0] = (acc_im0[v] - acc_n0[v]) * scale;
    A_re[row * 64 + c1] = acc_re1[v] * scale;
    A_im[row * 64 + c1] = (acc_im1[v] - acc_n1[v]) * scale;
  }
  __syncthreads();

  for (int idx = tid; idx < 4096; idx += 256) {
    V_re[idx] = ((idx >> 6) == (idx & 63)) ? 1.0f : 0.0f;
    V_im[idx] = 0.0f;
  }
  __syncthreads();

  // ---------------- Phase 2: parallel complex Jacobi ----------------
  const int pj  = tid >> 3;   // pair 0..31
  const int sub = tid & 7;    // 8 workers per pair

  for (int sw = 0; sw < NSWEEP; ++sw) {
    for (int r = 0; r < 63; ++r) {
      int p, q;
      if (pj == 0) { p = 63; q = r; }
      else { p = (r + pj) % 63; q = (r - pj + 63) % 63; }
      const int lo = (p < q) ? p : q;
      const int hi = (p < q) ? q : p;

      if (sub == 0) {
        const float are = A_re[lo * 64 + hi], aim = A_im[lo * 64 + hi];
        const float app = A_re[lo * 65],      aqq = A_re[hi * 65];
        const float rr  = sqrtf(are * are + aim * aim);
        float c = 1.0f, zr = 0.0f, zi = 0.0f;
        if (rr > 1e-20f) {
          const float tau = (aqq - app) / (2.0f * rr);
          const float tt  = (tau >= 0.0f ? 1.0f : -1.0f) /
                            (fabsf(tau) + sqrtf(1.0f + tau * tau));
          c = 1.0f / sqrtf(1.0f + tt * tt);
          const float s = tt * c;
          zr = s * (are / rr);
          zi = s * (aim / rr);
        }
        cP[pj] = c; zrP[pj] = zr; ziP[pj] = zi;
      }
      __syncthreads();
      const float c  = cP[pj];
      const float zr = zrP[pj];
      const float zi = ziP[pj];
      // column update: A(:,lo/hi), V(:,lo/hi)   (pairs own disjoint columns)
#pragma unroll
      for (int kb = 0; kb < 8; ++kb) {
        const int k = sub + kb * 8;
        float apr = A_re[k * 64 + lo], api = A_im[k * 64 + lo];
        float aqr = A_re[k * 64 + hi], aqi = A_im[k * 64 + hi];
        A_re[k * 64 + lo] = c * apr + (zr * aqr + zi * aqi);
        A_im[k * 64 + lo] = c * api + (zr * aqi - zi * aqr);
        A_re[k * 64 + hi] = c * aqr - (zr * apr - zi * api);
        A_im[k * 64 + hi] = c * aqi - (zr * api + zi * apr);
        float vpr = V_re[k * 64 + lo], vpi = V_im[k * 64 + lo];
        float vqr = V_re[k * 64 + hi], vqi = V_im[k * 64 + hi];
        V_re[k * 64 + lo] = c * vpr + (zr * vqr + zi * vqi);
        V_im[k * 64 + lo] = c * vpi + (zr * vqi - zi * vqr);
        V_re[k * 64 + hi] = c * vqr - (zr * vpr - zi * vpi);
        V_im[k * 64 + hi] = c * vqi - (zr * vpi + zi * vpr);
      }
      __syncthreads();
      // row update: A(lo/hi,:)   (pairs own disjoint rows)
#pragma unroll
      for (int kb = 0; kb < 8; ++kb) {
        const int k = sub + kb * 8;
        float apr = A_re[lo * 64 + k], api = A_im[lo * 64 + k];
        float aqr = A_re[hi * 64 + k], aqi = A_im[hi * 64 + k];
        A_re[lo * 64 + k] = c * apr + (zr * aqr - zi * aqi);
        A_im[lo * 64 + k] = c * api + (zr * aqi + zi * aqr);
        A_re[hi * 64 + k] = c * aqr - (zr * apr + zi * api);
        A_im[hi * 64 + k] = c * aqi - (zr * api - zi * apr);
      }
      __syncthreads();
    }
  }

  // ---------------- Phase 3: sort + outputs ----------------
  if (tid < 64) ev[tid] = A_re[tid * 65];
  __syncthreads();
  if (tid < 64) {
    const float mine = ev[tid];
    int rk = 0;
    for (int i = 0; i < 64; ++i) {
      const float o = ev[i];
      if (o > mine || (o == mine && i < tid)) rk++;
    }
    perm[rk] = tid;
  }
  __syncthreads();
  if (tid < 64) nrm[tid] = ev[perm[tid]] / ev[perm[0]];
  __syncthreads();

  for (int idx = tid; idx < 4096; idx += 256) {
    const int m = idx >> 6, r = idx & 63;
    const int col = perm[r];
    const float vr = V_re[m * 64 + col];
    const float vi = V_im[m * 64 + col];
    size_t o;
    if (r < K_SRC) o = SIG_OFF + (((size_t)b * 64 + m) * 8 + r) * 2;
    else           o = NOI_OFF + (((size_t)b * 64 + m) * 56 + (r - K_SRC)) * 2;
    out[o]     = vr;
    out[o + 1] = vi;
  }
  if (tid == 0) {
    int cnt = 0;
    for (int r2 = 0; r2 < 64; ++r2) cnt += (nrm[r2] > THRESH) ? 1 : 0;
    out[SRC_OFF + b] = (float)cnt;
    atomicAdd(&out[LEIG_OFF], (nrm[K_SRC - 1] - THRESH) * (nrm[K_SRC] - THRESH));
  }
}

extern "C" void kernel_launch(void* const* d_in, const int* in_sizes, int n_in,
                              void* d_out, int out_size, void* d_ws, size_t ws_size,
                              hipStream_t stream) {
  (void)in_sizes; (void)n_in; (void)out_size; (void)d_ws; (void)ws_size;
  const float* xre = (const float*)d_in[0];
  const float* xim = (const float*)d_in[1];
  float* out = (float*)d_out;

  leig_init_kernel<<<1, 32, 0, stream>>>(out);

  // dynamic LDS: A/V (4*4096) + params (288) floats = 66,688 bytes (< 320KB/WGP)
  const size_t shmem = (size_t)(4 * 4096 + 288) * sizeof(float);
  subspace_kernel<<<B_TOT, 256, shmem, stream>>>(xre, xim, out);
}